// VoxelResBackBone8xVoxelNeXtSPS_65000035058003
// MI455X (gfx1250) — compile-verified
//
#include <hip/hip_runtime.h>

// CDNA5 / gfx1250, wave32. Sparse submanifold conv backbone with
// V_WMMA_F32_16X16X4_F32 (full fp32 precision, matches fp32 reference).

typedef __attribute__((ext_vector_type(2))) float v2f;
typedef __attribute__((ext_vector_type(4))) float v4f;
typedef __attribute__((ext_vector_type(8))) float v8f;

#define NVOX 300000
#define KNBR 27
#define COUT 32
#define NTILES (NVOX / 16)   // 18750, exact

// ---------------------------------------------------------------------------
// Pack weights into WMMA B-fragment order, folding in the BN scale.
// Packed float index: (((k*KSTEPS + s)*2 + nt)*32 + lane)*2 + j
//   lane <16 : ci = 4s + j,     co = nt*16 + lane
//   lane >=16: ci = 4s + 2 + j, co = nt*16 + (lane-16)
// Also emits bias[co] = beta - mean*scale  (scale = gamma * rsqrt(var+eps)).
// ---------------------------------------------------------------------------
__global__ void pack_weights_kernel(const float* __restrict__ w,
                                    const float* __restrict__ bn,   // (4,32): gamma,beta,mean,var
                                    float* __restrict__ wpk,
                                    float* __restrict__ biasOut,
                                    int cin) {
    int t = blockIdx.x * blockDim.x + threadIdx.x;
    int ksteps = cin >> 2;
    int total  = KNBR * ksteps * 128;
    if (t < 32) {
        float gamma = bn[t], beta = bn[32 + t], mean = bn[64 + t], var = bn[96 + t];
        float scale = gamma * rsqrtf(var + 1e-3f);
        biasOut[t] = beta - mean * scale;
    }
    if (t >= total) return;
    int j    = t & 1;
    int lane = (t >> 1) & 31;
    int nt   = (t >> 6) & 1;
    int s    = (t >> 7) % ksteps;
    int k    = t / (ksteps * 128);
    int ci   = 4 * s + ((lane >= 16) ? 2 : 0) + j;
    int co   = nt * 16 + (lane & 15);
    float gamma = bn[co], var = bn[96 + co];
    float scale = gamma * rsqrtf(var + 1e-3f);
    wpk[t] = w[(k * cin + ci) * COUT + co] * scale;
}

// ---------------------------------------------------------------------------
// One subm-conv layer: out = epilogue( sum_k gather_k(fin) @ wpk[k] + bias )
// One wave = one 16-voxel tile; two 16x16 f32 accumulators cover cout=32.
// A gathered straight from global memory in WMMA A-fragment layout:
//   lane<16  : row = lane,    K-cols 4s,4s+1  (b64)
//   lane>=16 : row = lane-16, K-cols 4s+2,4s+3 (b64)
// ---------------------------------------------------------------------------
template <int CIN, bool HAS_ID>
__global__ __launch_bounds__(256) void subm_layer_kernel(
    const float* __restrict__ fin, const int* __restrict__ nbr,
    const float* __restrict__ wpk, const float* __restrict__ bias,
    const float* __restrict__ identity, float* __restrict__ fout,
    int ntiles) {
    constexpr int KSTEPS = CIN / 4;

    int wave = blockIdx.x * (blockDim.x >> 5) + (threadIdx.x >> 5);
    bool active = (wave < ntiles);
    if (!active) wave = ntiles - 1;   // keep whole wave alive: EXEC must be all-1s for WMMA
    int lane = threadIdx.x & 31;
    int row  = lane & 15;
    int hi   = lane >> 4;             // 0: K-cols {0,1}, 1: K-cols {2,3} within a K-step
    long long base = (long long)wave * 16;

    v8f acc0 = {0.f, 0.f, 0.f, 0.f, 0.f, 0.f, 0.f, 0.f};
    v8f acc1 = {0.f, 0.f, 0.f, 0.f, 0.f, 0.f, 0.f, 0.f};

    const int* nrow = nbr + (base + row) * KNBR;
    for (int k = 0; k < KNBR; ++k) {
        int idx = nrow[k];
        int safe = (idx < 0) ? 0 : idx;
        const float* arow = fin + (long long)safe * CIN + hi * 2;
        const float* bptr = wpk + (k * KSTEPS) * 128 + lane * 2;
#pragma unroll
        for (int s = 0; s < KSTEPS; ++s) {
            v2f a = *(const v2f*)(arow + 4 * s);
            if (idx < 0) { a.x = 0.f; a.y = 0.f; }   // v_cndmask, no EXEC change
            v2f b0 = *(const v2f*)(bptr + s * 128);
            v2f b1 = *(const v2f*)(bptr + s * 128 + 64);
            acc0 = __builtin_amdgcn_wmma_f32_16x16x4_f32(
                false, a, false, b0, (short)0, acc0, false, false);
            acc1 = __builtin_amdgcn_wmma_f32_16x16x4_f32(
                false, a, false, b1, (short)0, acc1, false, false);
        }
    }

    float bs0 = bias[row];
    float bs1 = bias[16 + row];
    if (active) {
        // C/D layout: VGPR v -> M = v + 8*hi, N = lane&15
#pragma unroll
        for (int v = 0; v < 8; ++v) {
            int m = v + hi * 8;
            long long o = (base + m) * COUT + row;
            float x0 = acc0[v] + bs0;
            float x1 = acc1[v] + bs1;
            if (HAS_ID) { x0 += identity[o]; x1 += identity[o + 16]; }
            fout[o]      = fmaxf(x0, 0.f);
            fout[o + 16] = fmaxf(x1, 0.f);
        }
    }
}

// ---------------------------------------------------------------------------
// voxel_importance = sigmoid(mean(|x|, axis=1))
// ---------------------------------------------------------------------------
__global__ void importance_kernel(const float* __restrict__ x,
                                  float* __restrict__ imp, int n) {
    int i = blockIdx.x * blockDim.x + threadIdx.x;
    if (i >= n) return;
    const v4f* p = (const v4f*)(x + (long long)i * COUT);
    float s = 0.f;
#pragma unroll
    for (int c = 0; c < 8; ++c) {
        v4f v = p[c];
        s += fabsf(v.x) + fabsf(v.y) + fabsf(v.z) + fabsf(v.w);
    }
    float m = s * (1.0f / 32.0f);
    imp[i] = 1.0f / (1.0f + expf(-m));
}

// ---------------------------------------------------------------------------
extern "C" void kernel_launch(void* const* d_in, const int* in_sizes, int n_in,
                              void* d_out, int out_size, void* d_ws, size_t ws_size,
                              hipStream_t stream) {
    const float* feat  = (const float*)d_in[0];
    const int*   nbr   = (const int*)d_in[1];
    const float* w_in  = (const float*)d_in[2];
    const float* bn_in = (const float*)d_in[3];
    const float* w1a   = (const float*)d_in[4];
    const float* bn1a  = (const float*)d_in[5];
    const float* w1b   = (const float*)d_in[6];
    const float* bn1b  = (const float*)d_in[7];
    const float* w2a   = (const float*)d_in[8];
    const float* bn2a  = (const float*)d_in[9];
    const float* w2b   = (const float*)d_in[10];
    const float* bn2b  = (const float*)d_in[11];

    // Workspace layout
    char* ws = (char*)d_ws;
    const size_t featBytes = (size_t)NVOX * COUT * sizeof(float);  // 38.4 MB
    float* bufA = (float*)(ws);
    float* bufB = (float*)(ws + featBytes);
    float* bufC = (float*)(ws + 2 * featBytes);
    float* wpk0 = (float*)(ws + 3 * featBytes);
    float* wpk1 = wpk0 + KNBR * 4 * 128;   // cin=16 packed: 13824 floats
    float* wpk2 = wpk1 + KNBR * 8 * 128;   // cin=32 packed: 27648 floats
    float* wpk3 = wpk2 + KNBR * 8 * 128;
    float* wpk4 = wpk3 + KNBR * 8 * 128;
    float* bias0 = wpk4 + KNBR * 8 * 128;
    float* bias1 = bias0 + 32;
    float* bias2 = bias1 + 32;
    float* bias3 = bias2 + 32;
    float* bias4 = bias3 + 32;

    // Pack weights (BN scale folded in), emit biases.
    {
        int tot16 = KNBR * 4 * 128;   // 13824
        int tot32 = KNBR * 8 * 128;   // 27648
        pack_weights_kernel<<<(tot16 + 255) / 256, 256, 0, stream>>>(w_in, bn_in, wpk0, bias0, 16);
        pack_weights_kernel<<<(tot32 + 255) / 256, 256, 0, stream>>>(w1a, bn1a, wpk1, bias1, 32);
        pack_weights_kernel<<<(tot32 + 255) / 256, 256, 0, stream>>>(w1b, bn1b, wpk2, bias2, 32);
        pack_weights_kernel<<<(tot32 + 255) / 256, 256, 0, stream>>>(w2a, bn2a, wpk3, bias3, 32);
        pack_weights_kernel<<<(tot32 + 255) / 256, 256, 0, stream>>>(w2b, bn2b, wpk4, bias4, 32);
    }

    // 8 waves / block, one tile per wave
    dim3 blk(256);
    dim3 grd((NTILES + 7) / 8);
    float* xout = (float*)d_out;

    subm_layer_kernel<16, false><<<grd, blk, 0, stream>>>(feat, nbr, wpk0, bias0, nullptr, bufA, NTILES);
    subm_layer_kernel<32, false><<<grd, blk, 0, stream>>>(bufA, nbr, wpk1, bias1, nullptr, bufB, NTILES);
    subm_layer_kernel<32, true ><<<grd, blk, 0, stream>>>(bufB, nbr, wpk2, bias2, bufA,    bufC, NTILES);
    subm_layer_kernel<32, false><<<grd, blk, 0, stream>>>(bufC, nbr, wpk3, bias3, nullptr, bufA, NTILES);
    subm_layer_kernel<32, true ><<<grd, blk, 0, stream>>>(bufA, nbr, wpk4, bias4, bufC,    xout, NTILES);

    importance_kernel<<<(NVOX + 255) / 256, 256, 0, stream>>>(xout, xout + (size_t)NVOX * COUT, NVOX);
}